// SMPLLoss_64072322121836
// MI455X (gfx1250) — compile-verified
//
#include <hip/hip_runtime.h>
#include <math.h>

// ---------------------------------------------------------------------------
// SMPL-style bilinear correlation loss for MI455X (gfx1250, wave32).
//
// Reference semantics:
//   gx = (gt_flow[:,0]+1)*(w-1)/2 ; gy = (gt_flow[:,1]+1)*(h-1)/2
//   fx = floor(gx), fy = floor(gy)
//   weights (per offset (0,0),(0,1),(1,0),(1,1)):
//     w0=(fy+1-gy)(fx+1-gx)  w1=(fy+1-gy)(gx-fx)
//     w2=(gy-fy)(fx+1-gx)    w3=(gy-fy)(gx-fx)
//   gathered_i[b,y,x] = corr_m[b, y*w+x, clip(fy+ox)*w + clip(fx+oy)]
//   loss = mean_i,b,y,x |gathered_i*vis - weights_i*vis|
//
// Design: one thread per (b,y,x) position (65536 threads). Coalesced loads of
// gt_flow/vis; 4 non-temporal scattered gathers from the 1 GiB corr tensor
// (we only touch ~0.1% of it -> NT hint avoids L2 pollution). Reduction:
// v_wmma_f32_16x16x4_f32 against an all-ones B matrix collapses the 32
// per-lane partials (placed in A VGPR0, A VGPR1 = 0) into row sums; summing
// the 8 D registers + one shfl_xor(16) yields the wave total in f32.
// Then LDS across 8 waves, one atomicAdd per block, finalize divides by N.
// ---------------------------------------------------------------------------

typedef __attribute__((ext_vector_type(2))) float v2f;
typedef __attribute__((ext_vector_type(8))) float v8f;

#define BB   16
#define HH   64
#define WW   64
#define NPOS (BB * HH * WW)        // 65536 positions
#define NTRM (BB * 4 * HH * WW)    // 262144 loss terms

__global__ void smpl_zero_ws(float* ws) { ws[0] = 0.0f; }

__global__ __launch_bounds__(256) void smpl_loss_kernel(
    const float* __restrict__ corr_m,   // [16, 4096, 4096]
    const float* __restrict__ gt_flow,  // [16, 2, 64, 64]
    const float* __restrict__ vis_mask, // [16, 1, 64, 64]
    float* __restrict__ ws)
{
    const int gid = blockIdx.x * blockDim.x + threadIdx.x;   // 0..65535
    const int b   = gid >> 12;      // gid / 4096
    const int p   = gid & 4095;     // y*64 + x

    // Coalesced parameter loads
    const float gxn = gt_flow[(size_t)(b * 2 + 0) * 4096 + p];
    const float gyn = gt_flow[(size_t)(b * 2 + 1) * 4096 + p];
    const float v   = vis_mask[(size_t)b * 4096 + p];

    const float gx = (gxn + 1.0f) * (0.5f * (WW - 1));
    const float gy = (gyn + 1.0f) * (0.5f * (HH - 1));
    const float fx = floorf(gx);
    const float fy = floorf(gy);

    const float wy0 = fy + 1.0f - gy;   // (fy+1-gy)
    const float wy1 = gy - fy;          // (gy-fy)
    const float wx0 = fx + 1.0f - gx;   // (fx+1-gx)
    const float wx1 = gx - fx;          // (gx-fx)
    const float wgt[4] = { wy0 * wx0, wy0 * wx1, wy1 * wx0, wy1 * wx1 };

    // clip(fy+{0,0,1,1}, 0, H-1) * W + clip(fx+{0,1,0,1}, 0, W-1)
    int y0 = (int)fy;     y0 = y0 < 0 ? 0 : (y0 > HH - 1 ? HH - 1 : y0);
    int y1 = (int)fy + 1; y1 = y1 < 0 ? 0 : (y1 > HH - 1 ? HH - 1 : y1);
    int x0 = (int)fx;     x0 = x0 < 0 ? 0 : (x0 > WW - 1 ? WW - 1 : x0);
    int x1 = (int)fx + 1; x1 = x1 < 0 ? 0 : (x1 > WW - 1 ? WW - 1 : x1);
    const int idx[4] = { y0 * WW + x0, y0 * WW + x1, y1 * WW + x0, y1 * WW + x1 };

    // Row corr_m[b, p, :] starts at element (b*4096 + p)*4096 == gid*4096
    const float* row = corr_m + (size_t)gid * 4096;

    float acc = 0.0f;
#pragma unroll
    for (int i = 0; i < 4; ++i) {
        const float g = __builtin_nontemporal_load(row + idx[i]);
        acc += fabsf(g * v - wgt[i] * v);
    }

    // ---- wave32 reduction ----
    float s;
#if __has_builtin(__builtin_amdgcn_wmma_f32_16x16x4_f32)
    // A (16x4 f32, 2 VGPRs/lane): VGPR0 = acc, VGPR1 = 0  -> each lane's acc
    // appears exactly once in A. B = all ones (4x16). D[m][n] = rowsum(A,m).
    // Sum of a lane's 8 D registers = total over 16 of the 32 accs; the
    // complementary half lives 16 lanes away -> one shfl_xor completes it.
    v2f a;  a[0] = acc;  a[1] = 0.0f;
    v2f bo; bo[0] = 1.0f; bo[1] = 1.0f;
    v8f c = {};
    c = __builtin_amdgcn_wmma_f32_16x16x4_f32(
            /*neg_a=*/false, a, /*neg_b=*/false, bo,
            /*c_mod=*/(short)0, c, /*reuse_a=*/false, /*reuse_b=*/false);
    s = c[0] + c[1] + c[2] + c[3] + c[4] + c[5] + c[6] + c[7];
    s += __shfl_xor(s, 16, 32);
#else
    s = acc;
    for (int d = 16; d > 0; d >>= 1) s += __shfl_xor(s, d, 32);
#endif

    // ---- cross-wave reduction (8 waves / block) ----
    __shared__ float lds[8];
    const int lane = threadIdx.x & 31;
    const int wave = threadIdx.x >> 5;
    if (lane == 0) lds[wave] = s;
    __syncthreads();
    if (threadIdx.x == 0) {
        float bs = 0.0f;
#pragma unroll
        for (int i = 0; i < 8; ++i) bs += lds[i];
        atomicAdd(ws, bs);
    }
}

__global__ void smpl_finalize(const float* __restrict__ ws,
                              float* __restrict__ out)
{
    out[0] = ws[0] * (1.0f / (float)NTRM);
}

extern "C" void kernel_launch(void* const* d_in, const int* in_sizes, int n_in,
                              void* d_out, int out_size, void* d_ws, size_t ws_size,
                              hipStream_t stream)
{
    const float* corr_m   = (const float*)d_in[0];
    const float* gt_flow  = (const float*)d_in[1];
    const float* vis_mask = (const float*)d_in[2];
    float*       out      = (float*)d_out;
    float*       ws       = (float*)d_ws;

    smpl_zero_ws<<<1, 1, 0, stream>>>(ws);
    smpl_loss_kernel<<<NPOS / 256, 256, 0, stream>>>(corr_m, gt_flow, vis_mask, ws);
    smpl_finalize<<<1, 1, 0, stream>>>(ws, out);
}